// RMSNorm_77309411566
// MI455X (gfx1250) — compile-verified
//
#include <hip/hip_runtime.h>
#include <hip/hip_fp16.h>
#include <stdint.h>

// RMSNorm + per-token symmetric int4 quantization (packed), fused.
//   x:        [T, H] fp32   (T = 8192, H = 4096)
//   out_q:    [T, H/2] int8 (low nibble = even idx, high nibble = odd idx)
//   scale:    [T] fp16      (max|normed| / 7)
//   input_sum:[T] fp16      (sum(normed))
//
// d_out layout assumption (outputs concatenated flat, native dtypes, return
// order): [ q : T*H/2 bytes ][ scale : T halves ][ input_sum : T halves ].

#define H          4096
#define HALF_H     (H / 2)
#define BLOCK      256
#define PER_THREAD 16          // floats staged per thread (4 x B128)
#define NWAVES     (BLOCK / 32)
#define EPS        1e-5f

__global__ __launch_bounds__(BLOCK) void rmsnorm_q4_kernel(
    const float* __restrict__ x,
    int8_t* __restrict__ out_q,
    __half* __restrict__ out_scale,
    __half* __restrict__ out_sum)
{
    __shared__ float row[H];                 // 16 KB staging for one token row
    __shared__ float wsum[NWAVES], wsqs[NWAVES], wmax[NWAVES];

    const int r = blockIdx.x;
    const int t = threadIdx.x;

    const float* src = x + (size_t)r * H + (size_t)t * PER_THREAD;

    // Low 32 bits of a generic pointer into __shared__ == LDS byte offset
    // (flat->LDS aperture mapping truncates to addr[31:0]).
    uint32_t lds_off = (uint32_t)(uintptr_t)(&row[t * PER_THREAD]);

    // Async-stage this thread's 64-byte slice. INST_OFFSET is added to BOTH
    // the LDS destination and the global source (CDNA5 ISA),
    // so one base VGPR (+pair) serves all four B128 copies. Tracked by
    // ASYNCcnt — CDNA5 async global->LDS path.
    asm volatile(
        "global_load_async_to_lds_b128 %0, %1, off\n\t"
        "global_load_async_to_lds_b128 %0, %1, off offset:16\n\t"
        "global_load_async_to_lds_b128 %0, %1, off offset:32\n\t"
        "global_load_async_to_lds_b128 %0, %1, off offset:48\n\t"
        "s_wait_asynccnt 0"
        :
        : "v"(lds_off), "v"(src)
        : "memory");
    __syncthreads();   // make every wave's staged data visible block-wide

    // ---- Pass over LDS: per-thread partial sum / sumsq / max|x| ----
    float4 v[4];
    const float4* lrow = (const float4*)(&row[t * PER_THREAD]);
    float s = 0.0f, ss = 0.0f, mx = 0.0f;
#pragma unroll
    for (int i = 0; i < 4; ++i) {
        float4 a = lrow[i];
        v[i] = a;
        s  += (a.x + a.y) + (a.z + a.w);
        ss += (a.x * a.x + a.y * a.y) + (a.z * a.z + a.w * a.w);
        mx  = fmaxf(mx, fmaxf(fmaxf(fabsf(a.x), fabsf(a.y)),
                              fmaxf(fabsf(a.z), fabsf(a.w))));
    }

    // ---- wave32 butterfly reduction (gfx1250 is wave32-only) ----
#pragma unroll
    for (int off = 16; off >= 1; off >>= 1) {
        s  += __shfl_xor(s,  off, 32);
        ss += __shfl_xor(ss, off, 32);
        mx  = fmaxf(mx, __shfl_xor(mx, off, 32));
    }

    const int wave = t >> 5;
    const int lane = t & 31;
    if (lane == 0) { wsum[wave] = s; wsqs[wave] = ss; wmax[wave] = mx; }
    __syncthreads();

    s = 0.0f; ss = 0.0f; mx = 0.0f;
#pragma unroll
    for (int w = 0; w < NWAVES; ++w) {
        s  += wsum[w];
        ss += wsqs[w];
        mx  = fmaxf(mx, wmax[w]);
    }

    const float rinv  = rsqrtf(ss * (1.0f / (float)H) + EPS);
    const float maxn  = mx * rinv;            // max|normed|
    const float scale = maxn / 7.0f;
    const float k     = 7.0f / mx;            // normed/scale == x * (7/max|x|)

    if (t == 0) {
        out_scale[r] = __float2half(scale);
        out_sum[r]   = __float2half(s * rinv);
    }

    // ---- Quantize 16 values -> 8 packed bytes, one B64 store per thread ----
    union { uint8_t b[8]; uint2 u; } pk;
    const float* vf = (const float*)v;
#pragma unroll
    for (int p = 0; p < 8; ++p) {
        float f0 = __builtin_rintf(vf[2 * p + 0] * k);   // round half-to-even
        float f1 = __builtin_rintf(vf[2 * p + 1] * k);
        f0 = fminf(fmaxf(f0, -8.0f), 7.0f);              // -> v_med3-class clamp
        f1 = fminf(fmaxf(f1, -8.0f), 7.0f);
        int q0 = (int)f0;
        int q1 = (int)f1;
        pk.b[p] = (uint8_t)((q0 & 0xF) | ((q1 & 0xF) << 4));
    }
    *(uint2*)(out_q + (size_t)r * HALF_H + (size_t)t * 8) = pk.u;
}

extern "C" void kernel_launch(void* const* d_in, const int* in_sizes, int n_in,
                              void* d_out, int out_size, void* d_ws, size_t ws_size,
                              hipStream_t stream) {
    (void)n_in; (void)d_ws; (void)ws_size; (void)out_size;

    const float* x = (const float*)d_in[0];
    const int T = in_sizes[0] / H;            // 8192 tokens

    int8_t* q      = (int8_t*)d_out;
    __half* scale  = (__half*)(q + (size_t)T * HALF_H);
    __half* in_sum = scale + T;

    rmsnorm_q4_kernel<<<T, BLOCK, 0, stream>>>(x, q, scale, in_sum);
}